// DilateAttention_1511828488301
// MI455X (gfx1250) — compile-verified
//
#include <hip/hip_runtime.h>

typedef __attribute__((ext_vector_type(2))) float v2f;
typedef __attribute__((ext_vector_type(4))) float v4f;
typedef __attribute__((ext_vector_type(8))) float v8f;

namespace {
constexpr int kB = 16, kC = 128, kH = 64, kW = 64;
constexpr int kHeads = 4, kHD = 32, kDil = 2, kKK = 9;
constexpr int kTile = 16;            // pixels per wave-tile (one row segment)
constexpr int kWaves = 8;            // waves per block
constexpr int kHW = kH * kW;         // channel plane stride (4096 floats)
constexpr float kScale = 0.17677669529663689f;  // 32^-0.5
constexpr int kTiles = kB * kHeads * kH * (kW / kTile);   // 16384
}

__global__ __launch_bounds__(kWaves * 32)
void dilate_attn_wmma(const float* __restrict__ q,
                      const float* __restrict__ kmat,
                      const float* __restrict__ vmat,
                      float* __restrict__ out)
{
  const int lane = threadIdx.x & 31;
  const int wave = threadIdx.x >> 5;
  const int tid  = blockIdx.x * kWaves + wave;

  const int xt = tid & 3;            // 4 column tiles per row
  const int y  = (tid >> 2) & 63;    // row
  const int h  = (tid >> 8) & 3;     // head
  const int b  = tid >> 10;          // batch
  const int x0 = xt * kTile;

  const int pix = lane & 15;         // pixel within tile (WMMA M/N index)
  const int hi  = lane >> 4;         // lane half

  const size_t chanBase = (size_t)(b * kC + h * kHD) * kHW;
  const float* qc = q    + chanBase;
  const float* kc = kmat + chanBase;
  const float* vc = vmat + chanBase;

  // ---- A fragments (Q) loaded straight from global in WMMA layout ----
  // 32-bit A 16x4: lanes 0-15 hold K=0,1 (vgpr 0,1); lanes 16-31 hold K=2,3.
  // Chained step s covers head-dims 4s..4s+3.
  v2f afrag[8];
  {
    const int qoff = y * kW + x0 + pix + (2 * hi) * kHW;
#pragma unroll
    for (int s = 0; s < 8; ++s) {
      afrag[s].x = qc[qoff + (4 * s + 0) * kHW];
      afrag[s].y = qc[qoff + (4 * s + 1) * kHW];
    }
  }

  // ---- phase 1: per-tap logits = diag( Q(16x32) x Kshift(32x16) ) ----
  float lgt[kKK];
#pragma unroll
  for (int t = 0; t < kKK; ++t) {
    const int dy = (t / 3 - 1) * kDil;
    const int dx = (t % 3 - 1) * kDil;
    const int yk = y + dy;
    const int xk = x0 + pix + dx;
    const bool ok = ((unsigned)yk < (unsigned)kH) && ((unsigned)xk < (unsigned)kW);

    float kf[16];
#pragma unroll
    for (int i = 0; i < 16; ++i) kf[i] = 0.f;   // zero padding (matches Unfold)
    if (ok) {
      const int koff = yk * kW + xk + (2 * hi) * kHW;
#pragma unroll
      for (int s = 0; s < 8; ++s) {
        kf[2 * s + 0] = kc[koff + (4 * s + 0) * kHW];
        kf[2 * s + 1] = kc[koff + (4 * s + 1) * kHW];
      }
    }

    v8f c = {0.f, 0.f, 0.f, 0.f, 0.f, 0.f, 0.f, 0.f};
#pragma unroll
    for (int s = 0; s < 8; ++s) {
      v2f bf;                     // B 4x16: column N=pix per lane, K across halves
      bf.x = kf[2 * s + 0];
      bf.y = kf[2 * s + 1];
      c = __builtin_amdgcn_wmma_f32_16x16x4_f32(false, afrag[s], false, bf,
                                                (short)0, c, false, false);
    }

    // diagonal D[p][p]: lane p (p<8) holds c[p]; lane p+16 (p>=8) holds c[p-8]
    const int rr = ((lane & 15) - 8 * hi) & 7;
    float d01 = (rr & 1) ? c[1] : c[0];
    float d23 = (rr & 1) ? c[3] : c[2];
    float d45 = (rr & 1) ? c[5] : c[4];
    float d67 = (rr & 1) ? c[7] : c[6];
    float dlo = (rr & 2) ? d23 : d01;
    float dhc = (rr & 2) ? d67 : d45;
    float dval = (rr & 4) ? dhc : dlo;

    // broadcast diag(pix) to every lane via one ds_bpermute (no LDS memory)
    const int src = pix + ((pix & 8) ? 16 : 0);
    lgt[t] = __int_as_float(
                 __builtin_amdgcn_ds_bpermute(src << 2, __float_as_int(dval))) *
             kScale;
  }

  // ---- softmax over 9 taps, fully in registers (redundant per lane pair) ----
  float w[kKK];
  float mx = lgt[0];
#pragma unroll
  for (int t = 1; t < kKK; ++t) mx = fmaxf(mx, lgt[t]);
  float sum = 0.f;
#pragma unroll
  for (int t = 0; t < kKK; ++t) { w[t] = __expf(lgt[t] - mx); sum += w[t]; }
  const float inv = 1.f / sum;
#pragma unroll
  for (int t = 0; t < kKK; ++t) w[t] *= inv;

  // ---- phase 2: out[pix, d] = sum_t w[t] * v[d, y+dy, x0+pix+dx] ----
  // lane owns pixel `pix`, head-dims [16*hi, 16*hi+16)
  float acc[16];
#pragma unroll
  for (int j = 0; j < 16; ++j) acc[j] = 0.f;

#pragma unroll
  for (int t = 0; t < kKK; ++t) {
    const int dy = (t / 3 - 1) * kDil;
    const int dx = (t % 3 - 1) * kDil;
    const int yk = y + dy;
    const int xk = x0 + pix + dx;
    const bool ok = ((unsigned)yk < (unsigned)kH) && ((unsigned)xk < (unsigned)kW);

    float vf[16];
#pragma unroll
    for (int j = 0; j < 16; ++j) vf[j] = 0.f;
    if (ok) {
      const int voff = yk * kW + xk + (16 * hi) * kHW;
#pragma unroll
      for (int j = 0; j < 16; ++j) vf[j] = vc[voff + j * kHW];
    }
#pragma unroll
    for (int j = 0; j < 16; ++j) acc[j] = fmaf(w[t], vf[j], acc[j]);
  }

  // ---- write out [B, H, W, 128]: 16 contiguous floats per lane, 64B aligned ----
  float* op = out + (((size_t)(b * kH + y) * kW + (x0 + pix)) * kC + h * kHD + hi * 16);
#pragma unroll
  for (int j = 0; j < 4; ++j) {
    v4f tv;
    tv.x = acc[4 * j + 0];
    tv.y = acc[4 * j + 1];
    tv.z = acc[4 * j + 2];
    tv.w = acc[4 * j + 3];
    *(v4f*)(op + 4 * j) = tv;
  }
}

extern "C" void kernel_launch(void* const* d_in, const int* in_sizes, int n_in,
                              void* d_out, int out_size, void* d_ws, size_t ws_size,
                              hipStream_t stream) {
  (void)in_sizes; (void)n_in; (void)d_ws; (void)ws_size; (void)out_size;
  const float* q = (const float*)d_in[0];
  const float* k = (const float*)d_in[1];
  const float* v = (const float*)d_in[2];
  float* o = (float*)d_out;

  const dim3 grid(kTiles / kWaves);       // 2048 blocks
  const dim3 block(kWaves * 32);          // 8 wave32 waves, fully independent
  hipLaunchKernelGGL(dilate_attn_wmma, grid, block, 0, stream, q, k, v, o);
}